// Solver_82033875353778
// MI455X (gfx1250) — compile-verified
//
#include <hip/hip_runtime.h>

// ---------------- problem constants (match reference) ----------------
constexpr int   NXc = 2048;
constexpr int   NVc = 1024;
constexpr float LXc   = 1.0f;
constexpr float VMAXc = 6.0f;
constexpr float DXc   = LXc / NXc;                 // 1/2048
constexpr float DVc   = 2.0f * VMAXc / NVc;        // 12/1024
constexpr float DTc   = 0.001f;
constexpr float T_LBOc = 1.0f;
constexpr float NU_EEc = 1.0f;
constexpr float NU_IIc = 0.1f;
// nu_shape params
constexpr float MIDc  = LXc / 3.0f;
constexpr float EFc   = LXc / 6.0f / 10.0f;        // LX/60

typedef float v2f __attribute__((ext_vector_type(2)));
typedef float v8f __attribute__((ext_vector_type(8)));

__device__ __forceinline__ float vs_of(int j) { return -VMAXc + DVc * (j + 0.5f); }
__device__ __forceinline__ float xs_of(int i) { return -0.5f * LXc + DXc * (i + 0.5f); }

__device__ __forceinline__ float minmodf(float a, float b) {
    return (a * b > 0.0f) ? copysignf(fminf(fabsf(a), fabsf(b)), a) : 0.0f;
}

// ---------------------------------------------------------------------
// Kernel 1: densities via WMMA.  ne[x] = DV * sum_v f[x,v].
// One wave (32 lanes) per 16 x-rows.  D = A(16xK chunk of f) * ones  ->
// every column of D is the row-sum of A; accumulate over K in f32.
// grid = (NX/16, 2 species), block = 32.
// ---------------------------------------------------------------------
__global__ void density_kernel(const float* __restrict__ fe,
                               const float* __restrict__ fi,
                               float* __restrict__ ne,
                               float* __restrict__ ni) {
    const float* f   = blockIdx.y ? fi : fe;
    float*       out = blockIdx.y ? ni : ne;
    const int x0   = blockIdx.x * 16;
    const int lane = threadIdx.x;          // 0..31, wave32
    const int m    = lane & 15;            // M row within tile
    const int half = lane >> 4;            // K pair selector

    const float* base = f + (size_t)(x0 + m) * NVc + 2 * half;

    v2f b; b.x = 1.0f; b.y = 1.0f;         // ones matrix (layout-agnostic)
    v8f c = {};                            // f32 accumulator, 16x16

    for (int k = 0; k < NVc; k += 4) {
        v2f a;
        a.x = base[k];                     // A[M=m][K=k+2*half]
        a.y = base[k + 1];                 // A[M=m][K=k+2*half+1]
        c = __builtin_amdgcn_wmma_f32_16x16x4_f32(
                /*neg_a=*/false, a, /*neg_b=*/false, b,
                /*c_mod=*/(short)0, c, /*reuse_a=*/false, /*reuse_b=*/false);
    }

    // D layout: VGPR j holds M=j (lanes 0..15) and M=j+8 (lanes 16..31),
    // N = lane % 16; all N columns are identical, so lanes with m==0 write.
    if (m == 0) {
        float* o = out + x0 + 8 * half;
#pragma unroll
        for (int j = 0; j < 8; ++j) o[j] = c[j] * DVc;
    }
}

// ---------------------------------------------------------------------
// Kernel 2: Poisson.  rho = ni - ne ; E = cumsum(rho)*DX ; E -= mean(E).
// One block of 1024 threads, each owns 2 consecutive cells.
// ---------------------------------------------------------------------
__global__ void poisson_kernel(const float* __restrict__ ne,
                               const float* __restrict__ ni,
                               float* __restrict__ E) {
    __shared__ float s[1024];
    const int t = threadIdx.x;
    const float a0 = ni[2 * t]     - ne[2 * t];
    const float a1 = ni[2 * t + 1] - ne[2 * t + 1];
    const float pair = a0 + a1;

    // inclusive Hillis-Steele scan of per-thread pair sums
    s[t] = pair;
    __syncthreads();
    for (int off = 1; off < 1024; off <<= 1) {
        float vv = (t >= off) ? s[t - off] : 0.0f;
        __syncthreads();
        s[t] += vv;
        __syncthreads();
    }
    const float incl = s[t];
    const float excl = incl - pair;
    const float e0 = (excl + a0) * DXc;
    const float e1 = (excl + a0 + a1) * DXc;

    // mean(E) over all 2048 cells
    __syncthreads();
    s[t] = e0 + e1;
    __syncthreads();
    for (int off = 512; off > 0; off >>= 1) {
        if (t < off) s[t] += s[t + off];
        __syncthreads();
    }
    const float mean = s[0] * (1.0f / NXc);
    E[2 * t]     = e0 - mean;
    E[2 * t + 1] = e1 - mean;
}

// ---------------------------------------------------------------------
// Kernel 3: fused MUSCL-minmod RHS + RK update.
//   out = alpha*f_extra + beta*f + gamma*(-v df/dx - Ec df/dv)
// x periodic (radius-2 wrap), v zero-ghosts (ghost f = 0 reproduces the
// reference's pad_v_zero exactly, including sigma at ghosts = 0).
// grid = (NV/256, NX), block = 256 (threads along contiguous v).
// ---------------------------------------------------------------------
__global__ void rhs_update_kernel(const float* __restrict__ f,
                                  const float* __restrict__ fextra,
                                  const float* __restrict__ E,
                                  float* __restrict__ out,
                                  float coefE, float alpha, float beta, float gamma) {
    const int v = blockIdx.x * blockDim.x + threadIdx.x;
    const int x = blockIdx.y;
    const size_t idx = (size_t)x * NVc + v;

    const int xm1 = (x + NXc - 1) & (NXc - 1);
    const int xm2 = (x + NXc - 2) & (NXc - 1);
    const int xp1 = (x + 1) & (NXc - 1);
    const int xp2 = (x + 2) & (NXc - 1);

    const float fc   = f[idx];
    const float fxm1 = f[(size_t)xm1 * NVc + v];
    const float fxm2 = f[(size_t)xm2 * NVc + v];
    const float fxp1 = f[(size_t)xp1 * NVc + v];
    const float fxp2 = f[(size_t)xp2 * NVc + v];

    const float fvm1 = (v >= 1)       ? f[idx - 1] : 0.0f;
    const float fvm2 = (v >= 2)       ? f[idx - 2] : 0.0f;
    const float fvp1 = (v <= NVc - 2) ? f[idx + 1] : 0.0f;
    const float fvp2 = (v <= NVc - 3) ? f[idx + 2] : 0.0f;

    // ---- x direction ----
    const float sxm1 = minmodf(fxm1 - fxm2, fc   - fxm1);
    const float sx   = minmodf(fc   - fxm1, fxp1 - fc);
    const float sxp1 = minmodf(fxp1 - fc,   fxp2 - fxp1);
    const float vj = vs_of(v);
    const float FxL = (vj > 0.0f) ? (fxm1 + 0.5f * sxm1) * vj : (fc   - 0.5f * sx)   * vj;
    const float FxR = (vj > 0.0f) ? (fc   + 0.5f * sx)   * vj : (fxp1 - 0.5f * sxp1) * vj;
    const float vdfdx = (FxR - FxL) * (1.0f / DXc);

    // ---- v direction ----
    const float svm1 = minmodf(fvm1 - fvm2, fc   - fvm1);
    const float sv   = minmodf(fc   - fvm1, fvp1 - fc);
    const float svp1 = minmodf(fvp1 - fc,   fvp2 - fvp1);
    const float Ec = coefE * E[x];
    const float FvL = (Ec > 0.0f) ? (fvm1 + 0.5f * svm1) * Ec : (fc   - 0.5f * sv)   * Ec;
    const float FvR = (Ec > 0.0f) ? (fc   + 0.5f * sv)   * Ec : (fvp1 - 0.5f * svp1) * Ec;
    const float Edfdv = (FvR - FvL) * (1.0f / DVc);

    const float rhs = -vdfdx - Edfdv;
    out[idx] = alpha * fextra[idx] + beta * fc + gamma * rhs;
}

// ---------------------------------------------------------------------
// Kernel 4: implicit LBO collision, Thomas solve per x-row (in place).
// One wave per row: coalesced stage into LDS, lane 0 runs the serial
// forward/backward recurrences in LDS, coalesced write-back.
// grid = NX/8, block = 256 (8 waves -> 8 rows), 64 KB LDS.
// ---------------------------------------------------------------------
__global__ void collide_kernel(float* __restrict__ f, float lam) {
    __shared__ float sb[8][NVc];   // b -> bp -> x
    __shared__ float scp[8][NVc];  // c'
    const int wave = threadIdx.x >> 5;
    const int lane = threadIdx.x & 31;
    const int row  = blockIdx.x * 8 + wave;
    float* frow = f + (size_t)row * NVc;

    for (int j = lane; j < NVc; j += 32) sb[wave][j] = frow[j];
    __syncthreads();

    if (lane == 0) {
        const float xs = xs_of(row);
        const float nu = 1.0f / (1.0f + 0.5f * (expf((xs - MIDc) / EFc) +
                                                expf((-xs - MIDc) / EFc)));
        const float c  = DTc * nu * lam;
        const float tD = T_LBOc / (DVc * DVc);
        const float dconst = 1.0f + 2.0f * c * tD;   // 1 - dt*nu*d, d = -2T/DV^2

        float cp = 0.0f, bp = 0.0f;
        for (int j = 0; j < NVc; ++j) {
            const float dl = (j == 0)       ? 0.0f : -c * (tD - vs_of(j - 1) * (0.5f / DVc));
            const float du = (j == NVc - 1) ? 0.0f : -c * (tD + vs_of(j + 1) * (0.5f / DVc));
            const float m   = dconst - dl * cp;
            const float inv = 1.0f / m;
            cp = du * inv;
            bp = (sb[wave][j] - dl * bp) * inv;
            scp[wave][j] = cp;
            sb[wave][j]  = bp;
        }
        float xn = 0.0f;
        for (int j = NVc - 1; j >= 0; --j) {
            xn = sb[wave][j] - scp[wave][j] * xn;
            sb[wave][j] = xn;
        }
    }
    __syncthreads();

    for (int j = lane; j < NVc; j += 32) frow[j] = sb[wave][j];
}

// ---------------------------------------------------------------------
extern "C" void kernel_launch(void* const* d_in, const int* in_sizes, int n_in,
                              void* d_out, int out_size, void* d_ws, size_t ws_size,
                              hipStream_t stream) {
    const float* fe = (const float*)d_in[0];
    const float* fi = (const float*)d_in[1];
    float* out = (float*)d_out;
    float* ws  = (float*)d_ws;

    const size_t NF = (size_t)NXc * NVc;
    float* fe1 = ws;            // stage-1 electron state
    float* fi1 = fe1 + NF;      // stage-1 ion state
    float* ne  = fi1 + NF;      // NX
    float* ni  = ne + NXc;      // NX
    float* Ef  = ni + NXc;      // NX

    float* fe2 = out;           // output species 0
    float* fi2 = out + NF;      // output species 1

    const float coefE_e = -1.0f;            // OMEGA_C_TAU * ZE / AE
    const float coefE_i = 1.0f / 1836.0f;   // OMEGA_C_TAU * ZI / AI

    dim3 gden(NXc / 16, 2);
    dim3 grhs(NVc / 256, NXc);

    // --- RK stage 1: f1 = f + dt * rhs(f) ---
    density_kernel<<<gden, 32, 0, stream>>>(fe, fi, ne, ni);
    poisson_kernel<<<1, 1024, 0, stream>>>(ne, ni, Ef);
    rhs_update_kernel<<<grhs, 256, 0, stream>>>(fe, fe, Ef, fe1, coefE_e, 0.0f, 1.0f, DTc);
    rhs_update_kernel<<<grhs, 256, 0, stream>>>(fi, fi, Ef, fi1, coefE_i, 0.0f, 1.0f, DTc);

    // --- RK stage 2: f2 = 0.5*(f + f1 + dt * rhs(f1)) ---
    density_kernel<<<gden, 32, 0, stream>>>(fe1, fi1, ne, ni);
    poisson_kernel<<<1, 1024, 0, stream>>>(ne, ni, Ef);
    rhs_update_kernel<<<grhs, 256, 0, stream>>>(fe1, fe, Ef, fe2, coefE_e, 0.5f, 0.5f, 0.5f * DTc);
    rhs_update_kernel<<<grhs, 256, 0, stream>>>(fi1, fi, Ef, fi2, coefE_i, 0.5f, 0.5f, 0.5f * DTc);

    // --- implicit collisions (in place on d_out) ---
    collide_kernel<<<NXc / 8, 256, 0, stream>>>(fe2, NU_EEc);
    collide_kernel<<<NXc / 8, 256, 0, stream>>>(fi2, NU_IIc);
}